// CopyOffsetDataset_6133213299103
// MI455X (gfx1250) — compile-verified
//
#include <hip/hip_runtime.h>
#include <stdint.h>

#define BB 64
#define CC 2048
#define VV 256
#define NROWS (BB * CC)          // 131072
#define PROB_W (VV + 1)          // 257
#define GROUPS (NROWS / 4)       // 32768 groups of 4 rows (4112 B, 16B-aligned)

// output layout (flat float32, reference return order)
#define OFF_KEY   0              // key_data: B*2
#define OFF_TOK   (BB * 2)       // tokens:   B*C
#define OFF_PROB  (OFF_TOK + NROWS)             // probs: B*C*257
#define OFF_IMASK (OFF_PROB + NROWS * PROB_W)   // input_mask
#define OFF_TMASK (OFF_IMASK + NROWS)           // target_mask

typedef unsigned int v4u __attribute__((ext_vector_type(4)));
typedef int          v8i __attribute__((ext_vector_type(8)));
typedef int          v4i __attribute__((ext_vector_type(4)));

// ---------------- threefry2x32 (JAX rotation schedule) ----------------
__device__ __forceinline__ void tf2x32(unsigned k0, unsigned k1,
                                       unsigned c0, unsigned c1,
                                       unsigned& o0, unsigned& o1) {
  unsigned ks2 = k0 ^ k1 ^ 0x1BD11BDAu;
  unsigned x0 = c0 + k0;
  unsigned x1 = c1 + k1;
#define TF_R(r) { x0 += x1; x1 = (x1 << (r)) | (x1 >> (32 - (r))); x1 ^= x0; }
  TF_R(13) TF_R(15) TF_R(26) TF_R(6)
  x0 += k1;  x1 += ks2 + 1u;
  TF_R(17) TF_R(29) TF_R(16) TF_R(24)
  x0 += ks2; x1 += k0 + 2u;
  TF_R(13) TF_R(15) TF_R(26) TF_R(6)
  x0 += k0;  x1 += k1 + 3u;
  TF_R(17) TF_R(29) TF_R(16) TF_R(24)
  x0 += k1;  x1 += ks2 + 4u;
  TF_R(13) TF_R(15) TF_R(26) TF_R(6)
  x0 += ks2; x1 += k0 + 5u;
#undef TF_R
  o0 = x0; o1 = x1;
}

__device__ __forceinline__ unsigned ballot32(bool p) {
  return (unsigned)__builtin_amdgcn_ballot_w32(p);
}

// -------- kernel 1: serial key chain per stream (the Amdahl part) --------
// JAX split(key,4): counts iota(8) -> halves (0..3),(4..7); reshape(4,2) gives
//   k1=(y0(0,4), y0(1,5))  k2=(y0(2,6), y0(3,7))
//   k3=(y1(0,4), y1(1,5))  next=(y1(2,6), y1(3,7))
__global__ void co_chain_kernel(const int* __restrict__ seeds,
                                unsigned* __restrict__ ws_keys,
                                float* __restrict__ out_keydata) {
  int b = threadIdx.x;
  if (b >= BB) return;
  unsigned k0 = 0u;                      // jax.random.key(int32 seed) -> (0, seed)
  unsigned k1 = (unsigned)seeds[b];
  out_keydata[2 * b + 0] = (float)k0;
  out_keydata[2 * b + 1] = (float)k1;
  for (int c = 0; c < CC; ++c) {
    size_t idx = (size_t)(b * CC + c) * 2;
    ws_keys[idx + 0] = k0;
    ws_keys[idx + 1] = k1;
    unsigned c0, c1, d0, d1;
    tf2x32(k0, k1, 2u, 6u, c0, c1);      // two independent hashes -> ILP 2
    tf2x32(k0, k1, 3u, 7u, d0, d1);
    k0 = c1; k1 = d1;                    // next_key
  }
}

// -------- kernel 2: all per-step random draws, fully parallel --------
__global__ void co_draws_kernel(const unsigned* __restrict__ ws_keys,
                                unsigned* __restrict__ ws_draws) {
  for (int i = blockIdx.x * blockDim.x + threadIdx.x; i < NROWS;
       i += gridDim.x * blockDim.x) {
    unsigned k0 = ws_keys[2 * i], k1 = ws_keys[2 * i + 1];
    unsigned a0, a1, b0, b1, c0, c1, d0, d1, t;
    tf2x32(k0, k1, 0u, 4u, a0, a1);
    tf2x32(k0, k1, 1u, 5u, b0, b1);
    tf2x32(k0, k1, 2u, 6u, c0, c1);
    tf2x32(k0, k1, 3u, 7u, d0, d1);
    unsigned bits1, bits2, bits3;
    tf2x32(a0, b0, 0u, 0u, bits1, t);    // k1 -> randint(0,V)
    tf2x32(c0, d0, 0u, 0u, bits2, t);    // k2 -> choice(I=10)
    tf2x32(a1, b1, 0u, 0u, bits3, t);    // k3 -> choice(arange(V))
    unsigned tok = bits1 & 255u;                       // span=256 -> bits&255
    unsigned hi = bits2 >> 16, lo = bits2 & 0xFFFFu;   // JAX randint span=10
    unsigned off10 = ((hi % 10u) * 6u + (lo % 10u)) % 10u;
    unsigned src = (off10 == 0u) ? 1u : 0u;
    unsigned off = bits3 & 255u;
    ws_draws[i] = tok | (off << 8) | (src << 16);
  }
}

// -------- kernel 3: 30-slot state machine, one wave32 per stream --------
__global__ void co_scan_kernel(const unsigned* __restrict__ ws_draws,
                               unsigned* __restrict__ ws_tok,
                               float* __restrict__ out_tokens,
                               float* __restrict__ out_imask,
                               float* __restrict__ out_tmask) {
  const int lane = threadIdx.x;
  const int b = blockIdx.x;
  const bool slotlane = lane < 30;
  const unsigned SLOTMASK = 0x3FFFFFFFu;
  int d = 0, w = 0, tg = 0;                     // per-lane slot state
  unsigned below = (lane == 0) ? 0u : ((1u << lane) - 1u);
  for (int c = 0; c < CC; ++c) {
    d -= 1;
    unsigned eq = ballot32(slotlane && d == 0);
    bool found = (eq != 0u);
    int slot = found ? (__ffs(eq) - 1) : 0;
    int tokS = __shfl(w, slot, 32);
    int tgS  = __shfl(tg, slot, 32);
    unsigned pk = ws_draws[b * CC + c];
    int rtok = (int)(pk & 255u);
    int roff = (int)((pk >> 8) & 255u);
    int rsrc = (int)((pk >> 16) & 1u);
    int token = found ? tokS : rtok;
    int is_target = found ? tgS : 0;
    bool is_source = rsrc && !found;
    unsigned occ = ballot32(slotlane && d >= roff && d <= roff + 2);  // isin
    bool use_source = is_source && (occ == 0u);
    unsigned freem = ballot32(slotlane && d < 0);
    int nfree = __popc(freem);
    int isfree = (int)((freem >> lane) & 1u);
    unsigned nonfree = (~freem) & SLOTMASK;
    // top_k((d<0),k=3): free slots ascending index first, then occupied
    int rank = isfree ? __popc(freem & below) : (nfree + __popc(nonfree & below));
    if (use_source && slotlane && rank < 3) {
      d = roff + rank;                                   // arange(3)+offset
      w = (rank == 0) ? roff : ((rank == 1) ? VV : token); // [offset, op, token]
      tg = (rank == 2) ? 1 : 0;                          // [F, F, T]
    }
    if (lane == 0) {
      int i = b * CC + c;
      out_tokens[i] = (float)token;
      out_imask[i]  = 1.0f;
      out_tmask[i]  = (float)is_target;
      ws_tok[i] = (unsigned)token | ((unsigned)is_target << 16);
    }
  }
}

// -------- kernel 4: 135 MB probs fill --------
// ntp staged to LDS via Tensor Data Mover; 4 rows per group = 257 aligned
// b128 stores (4112 B, 16B-aligned group pitch); prefetch of next token pack.
__global__ void co_probs_kernel(const float* __restrict__ ntp,
                                const unsigned* __restrict__ ws_tok,
                                float* __restrict__ probs) {
  __shared__ float lds_ntp[PROB_W];
#if __has_builtin(__builtin_amdgcn_tensor_load_to_lds)
  if (threadIdx.x < 32) {                    // single wave issues the TDM DMA
    unsigned lds_off = (unsigned)(uintptr_t)&lds_ntp[0];
    unsigned long long ga = (unsigned long long)(uintptr_t)ntp;
    v4u g0;
    g0.x = 1u;                                           // count=1 (valid D#)
    g0.y = lds_off;                                      // lds_addr
    g0.z = (unsigned)(ga & 0xFFFFFFFFu);                 // global_addr[31:0]
    g0.w = (unsigned)((ga >> 32) & 0x1FFFFFFu) | (2u << 30); // addr[56:32]|type=2
    v8i g1 = (v8i)0;
    g1[0] = (int)(2u << 16);                             // data_size=4B
    g1[1] = (int)(((unsigned)PROB_W & 0xFFFFu) << 16);   // tensor_dim0[15:0]
    g1[2] = (int)(((unsigned)PROB_W >> 16) | (1u << 16)); // dim0 hi | tensor_dim1=1
    g1[3] = (int)(((unsigned)PROB_W) << 16);             // tile_dim0=257
    g1[4] = 1;                                           // tile_dim1=1
    g1[5] = PROB_W;                                      // tensor_dim0_stride
    v4i gz = (v4i)0;
#if defined(__clang_major__) && (__clang_major__ >= 23)
    v8i gz8 = (v8i)0;
    __builtin_amdgcn_tensor_load_to_lds(g0, g1, gz, gz, gz8, 0);
#else
    __builtin_amdgcn_tensor_load_to_lds(g0, g1, gz, gz, 0);
#endif
    __builtin_amdgcn_s_wait_tensorcnt(0);
  }
#else
  for (int j = threadIdx.x; j < PROB_W; j += blockDim.x) lds_ntp[j] = ntp[j];
#endif
  __syncthreads();

  const int t = threadIdx.x;                 // blockDim.x == 256
  const uint4* tok4 = (const uint4*)ws_tok;
  for (int g = blockIdx.x; g < GROUPS; g += gridDim.x) {
    if (g + (int)gridDim.x < GROUPS)
      __builtin_prefetch(tok4 + g + gridDim.x, 0, 3);    // global_prefetch_b8
    uint4 p4 = tok4[g];
    unsigned pk[4] = {p4.x, p4.y, p4.z, p4.w};
    float* gp = probs + (size_t)g * 4 * PROB_W;          // 16B-aligned
    float4 v;
    float* vv = (float*)&v;
    int base = 4 * t;                                    // elems 4t..4t+3 of 1028
#pragma unroll
    for (int j = 0; j < 4; ++j) {
      int idx = base + j;
      int row = idx / PROB_W;                            // mul-high, 0..3
      int col = idx - row * PROB_W;
      unsigned p = pk[row];
      int tok = (int)(p & 0xFFFFu);
      int tgt = (int)((p >> 16) & 1u);
      vv[j] = tgt ? (col == tok ? 1.0f : 0.0f) : lds_ntp[col];
    }
    ((float4*)gp)[t] = v;                                // global_store_b128
    if (t == 0) {                                        // tail: elems 1024..1027
      float4 tv;
      float* tp = (float*)&tv;
#pragma unroll
      for (int j = 0; j < 4; ++j) {
        int idx = 1024 + j;                              // row 3, cols 253..256
        int col = idx - 3 * PROB_W;
        unsigned p = pk[3];
        int tok = (int)(p & 0xFFFFu);
        int tgt = (int)((p >> 16) & 1u);
        tp[j] = tgt ? (col == tok ? 1.0f : 0.0f) : lds_ntp[col];
      }
      *(float4*)(gp + 1024) = tv;
    }
  }
}

extern "C" void kernel_launch(void* const* d_in, const int* in_sizes, int n_in,
                              void* d_out, int out_size, void* d_ws, size_t ws_size,
                              hipStream_t stream) {
  (void)in_sizes; (void)n_in; (void)out_size; (void)ws_size;
  const int*   seeds = (const int*)d_in[0];     // [64] int32
  const float* ntp   = (const float*)d_in[1];   // [257] float32
  float* out = (float*)d_out;

  // workspace: keys (2*N u32) | draws (N u32) | tok (N u32)  = 2 MB
  unsigned* ws_keys  = (unsigned*)d_ws;
  unsigned* ws_draws = ws_keys + (size_t)2 * NROWS;
  unsigned* ws_tok   = ws_draws + (size_t)NROWS;   // 16B-aligned (offset 1.5 MB)

  co_chain_kernel<<<1, 64, 0, stream>>>(seeds, ws_keys, out + OFF_KEY);
  co_draws_kernel<<<512, 256, 0, stream>>>(ws_keys, ws_draws);
  co_scan_kernel<<<BB, 32, 0, stream>>>(ws_draws, ws_tok,
                                        out + OFF_TOK, out + OFF_IMASK,
                                        out + OFF_TMASK);
  co_probs_kernel<<<2048, 256, 0, stream>>>(ntp, ws_tok, out + OFF_PROB);
}